// NaiveReweightedLoss_29618094474146
// MI455X (gfx1250) — compile-verified
//
#include <hip/hip_runtime.h>
#include <hip/hip_bf16.h>
#include <math.h>

typedef __attribute__((ext_vector_type(2))) float v2f;
typedef __attribute__((ext_vector_type(8))) float v8f;

#define NROWS 32768
#define NCLS  1000
#define COL_TILES 63                 // ceil(1000/16)
#define COLS_PAD (COL_TILES * 16)    // 1008
#define WAVES_PER_BLOCK 8

#define LOG2E_F 1.4426950408889634f
#define LN2_F   0.6931471805599453f

// Fast stable softplus pieces using hardware v_exp_f32 / v_log_f32:
//   base(x) = log1p(exp(-|x|)) ~= ln2 * log2(1 + exp2(-|x|*log2e))
//   softplus(+x) = base + max(+x, 0) ; softplus(-x) = base + max(-x, 0)
__device__ __forceinline__ float softplus_base(float ap) {
    float e = __builtin_amdgcn_exp2f(-ap * LOG2E_F);   // v_exp_f32
    return LN2_F * __builtin_amdgcn_logf(1.0f + e);    // v_log_f32
}

// Kernel 1: per (16-column tile, row-chunk) wave, accumulate masked softplus
// sums and label counts per column via V_WMMA_F32_16X16X4_F32 with A == ones.
// With A all-ones, D[m][n] = sum_k B[k][n]; the K-slot assignment of loaded
// rows is irrelevant, making this robust to the exact B VGPR layout.
__global__ __launch_bounds__(256) void nrl_colreduce(
    const float* __restrict__ pred, const int* __restrict__ lab,
    float* __restrict__ ws, int rows_per_chunk)
{
    const int lane = threadIdx.x & 31;
    const int wave = threadIdx.x >> 5;
    const int tile = blockIdx.x * WAVES_PER_BLOCK + wave;
    if (tile >= COL_TILES) return;                 // wave-uniform exit (EXEC stays all-1s)

    const int chunk    = blockIdx.y;
    const int rchunks  = gridDim.y;
    const int col      = tile * 16 + (lane & 15);
    const bool colvalid = (col < NCLS);
    const int colc     = colvalid ? col : (NCLS - 1);   // clamp so loads stay in-bounds
    const int half     = lane >> 4;                     // 0: rows r,r+2 ; 1: rows r+1,r+3
    const int row0     = chunk * rows_per_chunk;

    v2f a; a[0] = 1.0f; a[1] = 1.0f;                    // ones A-matrix (16x4)
    v8f acc_sp = {0.f,0.f,0.f,0.f,0.f,0.f,0.f,0.f};    // sum softplus(-p) over pos
    v8f acc_sn = {0.f,0.f,0.f,0.f,0.f,0.f,0.f,0.f};    // sum softplus(+p) over neg
    v8f acc_np = {0.f,0.f,0.f,0.f,0.f,0.f,0.f,0.f};    // count pos
    v8f acc_nn = {0.f,0.f,0.f,0.f,0.f,0.f,0.f,0.f};    // count neg

#pragma unroll 2
    for (int r = 0; r < rows_per_chunk; r += 4) {
        const int rbase = row0 + r + half;
        const int i0 = rbase * NCLS + colc;
        const int i1 = i0 + 2 * NCLS;

        const float p0 = pred[i0];
        const float p1 = pred[i1];
        const int   t0 = lab[i0];
        const int   t1 = lab[i1];

        const float base0 = softplus_base(fabsf(p0));
        const float base1 = softplus_base(fabsf(p1));
        const float spm0 = base0 + fmaxf(-p0, 0.f);   // softplus(-p0)
        const float spp0 = base0 + fmaxf( p0, 0.f);   // softplus(+p0)
        const float spm1 = base1 + fmaxf(-p1, 0.f);
        const float spp1 = base1 + fmaxf( p1, 0.f);

        const bool pos0 = colvalid && (t0 == 1);
        const bool neg0 = colvalid && (t0 == 0);
        const bool pos1 = colvalid && (t1 == 1);
        const bool neg1 = colvalid && (t1 == 0);

        v2f bsp, bsn, bnp, bnn;
        bsp[0] = pos0 ? spm0 : 0.f;  bsp[1] = pos1 ? spm1 : 0.f;
        bsn[0] = neg0 ? spp0 : 0.f;  bsn[1] = neg1 ? spp1 : 0.f;
        bnp[0] = pos0 ? 1.f  : 0.f;  bnp[1] = pos1 ? 1.f  : 0.f;
        bnn[0] = neg0 ? 1.f  : 0.f;  bnn[1] = neg1 ? 1.f  : 0.f;

        acc_sp = __builtin_amdgcn_wmma_f32_16x16x4_f32(false, a, false, bsp, (short)0, acc_sp, false, false);
        acc_sn = __builtin_amdgcn_wmma_f32_16x16x4_f32(false, a, false, bsn, (short)0, acc_sn, false, false);
        acc_np = __builtin_amdgcn_wmma_f32_16x16x4_f32(false, a, false, bnp, (short)0, acc_np, false, false);
        acc_nn = __builtin_amdgcn_wmma_f32_16x16x4_f32(false, a, false, bnn, (short)0, acc_nn, false, false);
    }

    // D VGPR0: lanes 0-15 hold M=0,N=lane (all rows identical with ones-A),
    // so acc[0] on lanes 0-15 is the column sum for this tile.
    if (lane < 16 && colvalid) {
        const int A   = rchunks * COLS_PAD;
        const int idx = chunk * COLS_PAD + col;
        ws[idx]         = acc_sp[0];
        ws[A + idx]     = acc_sn[0];
        ws[2*A + idx]   = acc_np[0];
        ws[3*A + idx]   = acc_nn[0];
    }
}

// Kernel 2: deterministic fold of per-chunk partials -> per-class loss ->
// scalar mean over valid classes. One block, fixed-order tree reduction.
__global__ __launch_bounds__(256) void nrl_finalize(
    const float* __restrict__ ws, float* __restrict__ out, int rchunks)
{
    const int A = rchunks * COLS_PAD;
    const float* sp = ws;
    const float* sn = ws + A;
    const float* np = ws + 2 * A;
    const float* nn = ws + 3 * A;

    float loss = 0.f, cnt = 0.f;
    for (int c = threadIdx.x; c < NCLS; c += 256) {
        float s_sp = 0.f, s_sn = 0.f, s_np = 0.f, s_nn = 0.f;
        for (int k = 0; k < rchunks; ++k) {
            const int idx = k * COLS_PAD + c;
            s_sp += sp[idx];
            s_sn += sn[idx];
            s_np += np[idx];
            s_nn += nn[idx];
        }
        if (s_np > 0.f && s_nn > 0.f) {           // valid class
            loss += s_sp / s_np + s_sn / s_nn;    // max(n,1)==n when valid
            cnt  += 1.f;
        }
    }

    __shared__ float sl[256];
    __shared__ float sc[256];
    sl[threadIdx.x] = loss;
    sc[threadIdx.x] = cnt;
    __syncthreads();
    for (int s = 128; s > 0; s >>= 1) {
        if (threadIdx.x < s) {
            sl[threadIdx.x] += sl[threadIdx.x + s];
            sc[threadIdx.x] += sc[threadIdx.x + s];
        }
        __syncthreads();
    }
    if (threadIdx.x == 0)
        out[0] = sl[0] / fmaxf(sc[0], 1.f);
}

extern "C" void kernel_launch(void* const* d_in, const int* in_sizes, int n_in,
                              void* d_out, int out_size, void* d_ws, size_t ws_size,
                              hipStream_t stream) {
    const float* pred = (const float*)d_in[0];
    const int*   lab  = (const int*)d_in[1];
    float*       out  = (float*)d_out;
    float*       ws   = (float*)d_ws;

    // Pick row-chunk count to fit workspace: 4 float arrays of rchunks*COLS_PAD.
    int rchunks = 128;
    while (rchunks > 1 &&
           (size_t)rchunks * COLS_PAD * 4 * sizeof(float) > ws_size)
        rchunks >>= 1;
    const int rows_per_chunk = NROWS / rchunks;   // multiple of 4 for all powers of 2 <= 128

    dim3 grid((COL_TILES + WAVES_PER_BLOCK - 1) / WAVES_PER_BLOCK, rchunks);
    nrl_colreduce<<<grid, 256, 0, stream>>>(pred, lab, ws, rows_per_chunk);
    nrl_finalize<<<1, 256, 0, stream>>>(ws, out, rchunks);
}